// ConvG_4320737100475
// MI455X (gfx1250) — compile-verified
//
#include <hip/hip_runtime.h>
#include <math.h>

#define N_NODES   50000
#define N_EDGES   800000
#define N_FEAT    512
#define HIDDEN    256
#define N_CLASSES 40
#define K_HOPS    10

typedef __attribute__((ext_vector_type(2))) float v2f;
typedef __attribute__((ext_vector_type(8))) float v8f;

// ---------------------------------------------------------------------------
// GEMM1: h = relu(x @ W1 + b1)   [50000,512] x [512,256]
// One wave32 per 16x16 output tile, K stepped by 4 via V_WMMA_F32_16X16X4_F32.
// A (16x4 f32) per-lane: lanes 0-15 -> M=lane, K = k0+0/k0+1 in v0/v1;
//                        lanes 16-31 -> M=lane-16, K = k0+2/k0+3. (ISA 7.12.2)
// B (4x16) mirrors A: lane half selects K pair, lane&15 selects N column.
// C/D: VGPR g -> M = g + 8*half, N = lane&15.
// ---------------------------------------------------------------------------
__global__ __launch_bounds__(256) void gemm1_relu_wmma(
    const float* __restrict__ x, const float* __restrict__ W1,
    const float* __restrict__ b1, float* __restrict__ h)
{
    const int warp = threadIdx.x >> 5;
    const int lane = threadIdx.x & 31;
    const int tile = blockIdx.x * 8 + warp;       // 0..49999 exact (grid=6250)
    const int tm   = tile >> 4;                   // 0..3124
    const int tn   = tile & 15;                   // 0..15
    const int half = lane >> 4;
    const int l    = lane & 15;

    const int row = tm * 16 + l;                  // always < 50000
    const int col = tn * 16 + l;

    const float* arow = x  + (size_t)row * N_FEAT + half * 2;
    const float* bcol = W1 + (size_t)(half * 2) * HIDDEN + col;

    v8f c = {};
    for (int k0 = 0; k0 < N_FEAT; k0 += 4) {
        v2f a = *(const v2f*)(arow + k0);
        v2f b;
        b.x = bcol[(size_t)k0 * HIDDEN];
        b.y = bcol[(size_t)(k0 + 1) * HIDDEN];
        c = __builtin_amdgcn_wmma_f32_16x16x4_f32(
                false, a, false, b, (short)0, c, false, false);
    }

    const float bias = b1[col];
#pragma unroll
    for (int g = 0; g < 8; ++g) {
        int m = tm * 16 + half * 8 + g;
        float v = c[g] + bias;
        h[(size_t)m * HIDDEN + col] = v > 0.0f ? v : 0.0f;
    }
}

// ---------------------------------------------------------------------------
// GEMM2: logits = out @ W2 + b2   [50000,256] x [256,40] (cols padded to 48)
// 3 waves per block (one M row of tiles), exact grid -> EXEC stays full;
// out-of-range columns handled with selects, not branches, inside the K loop.
// ---------------------------------------------------------------------------
__global__ __launch_bounds__(96) void gemm2_wmma(
    const float* __restrict__ out_acc, const float* __restrict__ W2,
    const float* __restrict__ b2, float* __restrict__ logits)
{
    const int tn   = threadIdx.x >> 5;            // 0..2 (N tile)
    const int lane = threadIdx.x & 31;
    const int tm   = blockIdx.x;                  // 0..3124 (M tile)
    const int half = lane >> 4;
    const int l    = lane & 15;

    const int row   = tm * 16 + l;
    const int col   = tn * 16 + l;
    const bool colok = col < N_CLASSES;

    const float* arow = out_acc + (size_t)row * HIDDEN + half * 2;

    v8f c = {};
    for (int k0 = 0; k0 < HIDDEN; k0 += 4) {
        v2f a = *(const v2f*)(arow + k0);
        int kb = k0 + half * 2;
        v2f b;
        b.x = colok ? W2[(size_t)kb * N_CLASSES + col]       : 0.0f;
        b.y = colok ? W2[(size_t)(kb + 1) * N_CLASSES + col] : 0.0f;
        c = __builtin_amdgcn_wmma_f32_16x16x4_f32(
                false, a, false, b, (short)0, c, false, false);
    }

    if (colok) {
        const float bias = b2[col];
#pragma unroll
        for (int g = 0; g < 8; ++g) {
            int m = tm * 16 + half * 8 + g;
            logits[(size_t)m * N_CLASSES + col] = c[g] + bias;
        }
    }
}

// --------------------------- degree / normalization ------------------------
__global__ void init_deg_kernel(float* __restrict__ deg)
{
    int i = blockIdx.x * blockDim.x + threadIdx.x;
    if (i < N_NODES) deg[i] = 1.0f;               // self-loop
}

__global__ void accum_deg_kernel(const int* __restrict__ dst,
                                 float* __restrict__ deg)
{
    int e = blockIdx.x * blockDim.x + threadIdx.x; // grid exact: 800000/256
    atomicAdd(&deg[dst[e]], 1.0f);
}

__global__ void finish_deg_kernel(const float* __restrict__ deg,
                                  float* __restrict__ dinv,
                                  float* __restrict__ selfw)
{
    int i = blockIdx.x * blockDim.x + threadIdx.x;
    if (i < N_NODES) {
        float d = rsqrtf(deg[i]);
        dinv[i]  = d;
        selfw[i] = d * d;
    }
}

__global__ void edge_norm_kernel(const int* __restrict__ src,
                                 const int* __restrict__ dst,
                                 const float* __restrict__ dinv,
                                 float* __restrict__ norm)
{
    int e = blockIdx.x * blockDim.x + threadIdx.x; // grid exact
    norm[e] = dinv[src[e]] * dinv[dst[e]];
}

// ------------------------------- propagation -------------------------------
// out = temp[0] * h
__global__ void scale_init_kernel(const float* __restrict__ h,
                                  float* __restrict__ out,
                                  const float* __restrict__ temp)
{
    int i = blockIdx.x * blockDim.x + threadIdx.x; // grid exact: 12.8M/256
    out[i] = temp[0] * h[i];
}

// h_next = self_w[node] * h   (HIDDEN = 256 -> node = i >> 8)
__global__ void selfinit_kernel(const float* __restrict__ h,
                                const float* __restrict__ selfw,
                                float* __restrict__ hn)
{
    int i = blockIdx.x * blockDim.x + threadIdx.x;
    hn[i] = selfw[i >> 8] * h[i];
}

// Scatter: 64 consecutive threads handle one edge (float4 per thread) so the
// h[src] row read is fully coalesced and the 4 atomics per thread land in one
// contiguous 1KB h[dst] row (L2-friendly on the 192MB L2).
__global__ void scatter_kernel(const float* __restrict__ h_src,
                               float* __restrict__ h_dst,
                               const int* __restrict__ src,
                               const int* __restrict__ dst,
                               const float* __restrict__ norm)
{
    unsigned tid = blockIdx.x * blockDim.x + threadIdx.x; // grid exact: 51.2M/256
    int e = tid >> 6;            // edge index
    int g = tid & 63;            // 4-float group within the 256-wide row
    int s = src[e];
    int d = dst[e];
    float w = norm[e];
    float4 v = *((const float4*)(h_src + (size_t)s * HIDDEN) + g);
    float* p = h_dst + (size_t)d * HIDDEN + g * 4;
    atomicAdd(p + 0, w * v.x);
    atomicAdd(p + 1, w * v.y);
    atomicAdd(p + 2, w * v.z);
    atomicAdd(p + 3, w * v.w);
}

// out += temp[k] * h_next
__global__ void accum_kernel(const float* __restrict__ hn,
                             float* __restrict__ out,
                             const float* __restrict__ temp, int k)
{
    int i = blockIdx.x * blockDim.x + threadIdx.x;
    out[i] += temp[k] * hn[i];
}

// ------------------------------- log-softmax -------------------------------
__global__ void log_softmax_kernel(float* __restrict__ logits)
{
    int r = blockIdx.x * blockDim.x + threadIdx.x;
    if (r >= N_NODES) return;
    float* p = logits + (size_t)r * N_CLASSES;
    float mx = -__builtin_inff();
    for (int j = 0; j < N_CLASSES; ++j) mx = fmaxf(mx, p[j]);
    float s = 0.0f;
    for (int j = 0; j < N_CLASSES; ++j) s += expf(p[j] - mx);
    float lg = logf(s);
    for (int j = 0; j < N_CLASSES; ++j) p[j] = p[j] - mx - lg;
}

// ---------------------------------------------------------------------------
extern "C" void kernel_launch(void* const* d_in, const int* in_sizes, int n_in,
                              void* d_out, int out_size, void* d_ws, size_t ws_size,
                              hipStream_t stream)
{
    const float* x    = (const float*)d_in[0];
    const float* W1   = (const float*)d_in[1];
    const float* b1   = (const float*)d_in[2];
    const float* W2   = (const float*)d_in[3];
    const float* b2   = (const float*)d_in[4];
    const float* temp = (const float*)d_in[5];
    const int*   ei   = (const int*)d_in[6];      // [2, E] int32 (JAX x32)
    const int*   src  = ei;
    const int*   dst  = ei + N_EDGES;

    float* logits = (float*)d_out;                // [50000, 40]

    // workspace carve-up (256B aligned)
    char* base = (char*)d_ws;
    size_t off = 0;
    auto carve = [&](size_t bytes) -> char* {
        char* p = base + off;
        off += (bytes + 255) & ~(size_t)255;
        return p;
    };
    const size_t NH = (size_t)N_NODES * HIDDEN * sizeof(float); // 51.2 MB
    float* hA      = (float*)carve(NH);
    float* hB      = (float*)carve(NH);
    float* out_acc = (float*)carve(NH);
    float* deg     = (float*)carve((size_t)N_NODES * sizeof(float));
    float* dinv    = (float*)carve((size_t)N_NODES * sizeof(float));
    float* selfw   = (float*)carve((size_t)N_NODES * sizeof(float));
    float* norm    = (float*)carve((size_t)N_EDGES * sizeof(float));
    (void)ws_size; (void)in_sizes; (void)n_in; (void)out_size;

    const int NHT = N_NODES * HIDDEN;             // 12,800,000

    // 1) feature transform (WMMA f32)
    gemm1_relu_wmma<<<(N_NODES / 16) * 16 / 8, 256, 0, stream>>>(x, W1, b1, hA);

    // 2) degrees + edge norms
    init_deg_kernel<<<(N_NODES + 255) / 256, 256, 0, stream>>>(deg);
    accum_deg_kernel<<<N_EDGES / 256, 256, 0, stream>>>(dst, deg);
    finish_deg_kernel<<<(N_NODES + 255) / 256, 256, 0, stream>>>(deg, dinv, selfw);
    edge_norm_kernel<<<N_EDGES / 256, 256, 0, stream>>>(src, dst, dinv, norm);

    // 3) out = temp[0] * h
    scale_init_kernel<<<NHT / 256, 256, 0, stream>>>(hA, out_acc, temp);

    // 4) K hops of A_hat propagation
    float* hcur = hA;
    float* hnext = hB;
    for (int k = 1; k <= K_HOPS; ++k) {
        selfinit_kernel<<<NHT / 256, 256, 0, stream>>>(hcur, selfw, hnext);
        scatter_kernel<<<(N_EDGES * 64u) / 256, 256, 0, stream>>>(hcur, hnext,
                                                                  src, dst, norm);
        accum_kernel<<<NHT / 256, 256, 0, stream>>>(hnext, out_acc, temp, k);
        float* t = hcur; hcur = hnext; hnext = t;
    }

    // 5) classifier GEMM (WMMA f32) + log-softmax in place
    gemm2_wmma<<<N_NODES / 16, 96, 0, stream>>>(out_acc, W2, b2, logits);
    log_softmax_kernel<<<(N_NODES + 255) / 256, 256, 0, stream>>>(logits);
}